// DataDependentRBFKernel_46566035423346
// MI455X (gfx1250) — compile-verified
//
#include <hip/hip_runtime.h>
#include <hip/hip_bf16.h>
#include <math.h>

// ---------------------------------------------------------------------------
// Problem dims (fixed by the reference setup_inputs)
// ---------------------------------------------------------------------------
#define B_DIM   8
#define N_DIM   1024
#define R_TOTAL (B_DIM * N_DIM)   // 8192 rows (b,n)
#define E_DIM   256               // embedding dim
#define H1      32                // hidden 1
#define H2      16                // hidden 2
#define M_DIM   2048              // number of z points

typedef __attribute__((ext_vector_type(2))) float v2f;
typedef __attribute__((ext_vector_type(4))) float v4f;
typedef __attribute__((ext_vector_type(8))) float v8f;

__device__ __forceinline__ float gelu_exact(float x) {
  // matches jax.nn.gelu(approximate=False)
  return 0.5f * x * (1.0f + erff(x * 0.70710678118654752440f));
}

// ---------------------------------------------------------------------------
// Kernel 1: sigma MLP.  One wave (32 lanes) handles 16 rows of embeddings.
// Layer 1 (256 -> 32) runs on V_WMMA_F32_16X16X4_F32: two 16x16 col tiles,
// K stepped by 4 => 2 * 64 = 128 WMMA ops per wave. Layers 2/3 are scalar.
// Writes inv2s2[r] = 1 / (2*sigma^2) into workspace.
// ---------------------------------------------------------------------------
__global__ __launch_bounds__(32) void sigma_mlp_wmma_kernel(
    const float* __restrict__ emb,   // [R_TOTAL, E_DIM]
    const float* __restrict__ w1,    // [E_DIM, H1]
    const float* __restrict__ b1,    // [H1]
    const float* __restrict__ w2,    // [H1, H2]
    const float* __restrict__ b2,    // [H2]
    const float* __restrict__ w3,    // [H2]
    const float* __restrict__ b3,    // [1]
    float* __restrict__ inv2s2)      // [R_TOTAL]
{
  __shared__ float h_lds[16][H1 + 1];   // +1 pad: conflict-free column reads

  const int lane  = threadIdx.x;        // 0..31 (wave32)
  const int mrow  = lane & 15;          // M index within tile / N index of B
  const int khalf = lane >> 4;          // 0: K pair {0,1}; 1: K pair {2,3}
  const int row0  = blockIdx.x * 16;

  const float* arow = emb + (size_t)(row0 + mrow) * E_DIM;

  v8f acc0 = {};   // D tile, columns 0..15
  v8f acc1 = {};   // D tile, columns 16..31

  for (int k = 0; k < E_DIM; k += 4) {
    const int krow = k + 2 * khalf;
    // A 16x4 f32 layout: v0 = A[M][krow], v1 = A[M][krow+1]
    v2f a = *(const v2f*)(arow + krow);
    // B 4x16 f32 layout (symmetric): v0 = B[krow][col], v1 = B[krow+1][col]
    v2f bt0, bt1;
    bt0.x = w1[(size_t)krow * H1 + mrow];
    bt0.y = w1[(size_t)(krow + 1) * H1 + mrow];
    bt1.x = w1[(size_t)krow * H1 + 16 + mrow];
    bt1.y = w1[(size_t)(krow + 1) * H1 + 16 + mrow];

    acc0 = __builtin_amdgcn_wmma_f32_16x16x4_f32(
        /*neg_a=*/false, a, /*neg_b=*/false, bt0,
        /*c_mod=*/(short)0, acc0, /*reuse_a=*/false, /*reuse_b=*/false);
    acc1 = __builtin_amdgcn_wmma_f32_16x16x4_f32(
        false, a, false, bt1, (short)0, acc1, false, false);
  }

  // Bias + exact GELU, scatter D layout (lane holds col=mrow, rows r+8*khalf)
  const float bias0 = b1[mrow];
  const float bias1 = b1[16 + mrow];
#pragma unroll
  for (int r = 0; r < 8; ++r) {
    const int mr = r + 8 * khalf;
    h_lds[mr][mrow]      = gelu_exact(acc0[r] + bias0);
    h_lds[mr][16 + mrow] = gelu_exact(acc1[r] + bias1);
  }
  __syncthreads();

  // Layers 2 & 3: lanes 0..15 each own one row (528 MACs -- negligible)
  if (lane < 16) {
    const int mr = lane;
    float dot3 = 0.0f;
#pragma unroll
    for (int j = 0; j < H2; ++j) {
      float t = b2[j];
#pragma unroll
      for (int i = 0; i < H1; ++i) {
        t = fmaf(h_lds[mr][i], w2[i * H2 + j], t);
      }
      dot3 = fmaf(gelu_exact(t), w3[j], dot3);
    }
    dot3 += b3[0];
    const float s     = 1.0f / (1.0f + __expf(-dot3));        // sigmoid
    const float sigma = 0.1f + 9.9f * s;                      // [0.1, 10]
    inv2s2[row0 + mr] = 1.0f / (2.0f * sigma * sigma);
  }
}

// ---------------------------------------------------------------------------
// Kernel 2: RBF evaluation, store-bandwidth bound (64 MiB out).
// Each block: z staged in LDS (16 KB), 4 rows x 2048 m, b128 NT stores.
// ---------------------------------------------------------------------------
#define ROWS_PER_BLOCK 4

__global__ __launch_bounds__(256) void rbf_stream_kernel(
    const float* __restrict__ z,       // [M_DIM, 2]
    const float* __restrict__ mu,      // [R_TOTAL, 2]
    const float* __restrict__ inv2s2,  // [R_TOTAL]
    float* __restrict__ out)           // [R_TOTAL, M_DIM]
{
  __shared__ float s_zx[M_DIM];
  __shared__ float s_zy[M_DIM];

  const int tid = threadIdx.x;

  // Stage z: 2048 float2 pairs -> 1024 b128 loads across 256 threads
  for (int i = tid; i < M_DIM / 2; i += 256) {
    v4f v = ((const v4f*)z)[i];
    s_zx[2 * i]     = v.x;
    s_zy[2 * i]     = v.y;
    s_zx[2 * i + 1] = v.z;
    s_zy[2 * i + 1] = v.w;
  }
  __syncthreads();

  const int r0 = blockIdx.x * ROWS_PER_BLOCK;
#pragma unroll
  for (int rr = 0; rr < ROWS_PER_BLOCK; ++rr) {
    const int r   = r0 + rr;
    const float mx = mu[2 * r];
    const float my = mu[2 * r + 1];
    const float c  = inv2s2[r];
    float* orow = out + (size_t)r * M_DIM;

#pragma unroll
    for (int it = 0; it < M_DIM / (256 * 4); ++it) {
      const int m = (it * 256 + tid) * 4;
      v4f o;
      {
        float dx = s_zx[m]     - mx, dy = s_zy[m]     - my;
        o.x = __expf(-(dx * dx + dy * dy) * c);
      }
      {
        float dx = s_zx[m + 1] - mx, dy = s_zy[m + 1] - my;
        o.y = __expf(-(dx * dx + dy * dy) * c);
      }
      {
        float dx = s_zx[m + 2] - mx, dy = s_zy[m + 2] - my;
        o.z = __expf(-(dx * dx + dy * dy) * c);
      }
      {
        float dx = s_zx[m + 3] - mx, dy = s_zy[m + 3] - my;
        o.w = __expf(-(dx * dx + dy * dy) * c);
      }
      // Output is written once and never re-read on device: non-temporal b128
      __builtin_nontemporal_store(o, (v4f*)(orow + m));
    }
  }
}

// ---------------------------------------------------------------------------
// Launch: inputs in setup_inputs() order: z, mu, embeddings, w1,b1,w2,b2,w3,b3
// ---------------------------------------------------------------------------
extern "C" void kernel_launch(void* const* d_in, const int* in_sizes, int n_in,
                              void* d_out, int out_size, void* d_ws, size_t ws_size,
                              hipStream_t stream) {
  const float* z   = (const float*)d_in[0];
  const float* mu  = (const float*)d_in[1];
  const float* emb = (const float*)d_in[2];
  const float* w1  = (const float*)d_in[3];
  const float* b1  = (const float*)d_in[4];
  const float* w2  = (const float*)d_in[5];
  const float* b2  = (const float*)d_in[6];
  const float* w3  = (const float*)d_in[7];
  const float* b3  = (const float*)d_in[8];
  float* out    = (float*)d_out;
  float* invbuf = (float*)d_ws;   // 8192 floats of scratch

  sigma_mlp_wmma_kernel<<<R_TOTAL / 16, 32, 0, stream>>>(
      emb, w1, b1, w2, b2, w3, b3, invbuf);

  rbf_stream_kernel<<<R_TOTAL / ROWS_PER_BLOCK, 256, 0, stream>>>(
      z, mu, invbuf, out);
}